// Encoder_43052752175741
// MI455X (gfx1250) — compile-verified
//
#include <hip/hip_runtime.h>
#include <stdint.h>

// ---------------------------------------------------------------------------
// Problem constants (from the reference)
// ---------------------------------------------------------------------------
#define FEAT_DIM   512
#define EMBED_DIM  512
#define BATCH      16384
#define NUM_SAMPLE 32
#define K_DIM      (2 * FEAT_DIM)     // 1024

// GEMM tiling
#define BM 128                        // block tile M (embed rows)
#define BN 128                        // block tile N (batch cols)
#define KC 16                         // K chunk staged in LDS per buffer
#define NCHUNK (K_DIM / KC)           // 64
#define LDS_STRIDE 20                 // floats per LDS row: 80B, keeps b128 dst
                                      // 16B-aligned and ds_load_b64 conflict-free

typedef __attribute__((ext_vector_type(2))) float v2f;
typedef __attribute__((ext_vector_type(8))) float v8f;

// ---------------------------------------------------------------------------
// Kernel 1: combined[b, 0:512]    = features[nodes[b]]
//           combined[b, 512:1024] = mean_j features[neigh_idx[b, j]]
// One block per batch row, 256 threads, float2 per thread (fully coalesced).
// ---------------------------------------------------------------------------
__global__ void __launch_bounds__(256)
agg_combined_kernel(const float* __restrict__ features,
                    const int*   __restrict__ nodes,
                    const int*   __restrict__ neigh,
                    float*       __restrict__ combined) {
    const int b = blockIdx.x;
    const int t = threadIdx.x;            // 0..255 -> columns 2t, 2t+1

    const int node = nodes[b];            // uniform -> scalar load
    const float2 selfv =
        *(const float2*)(features + (size_t)node * FEAT_DIM + 2 * t);

    float2 acc; acc.x = 0.f; acc.y = 0.f;
    const int* nrow = neigh + (size_t)b * NUM_SAMPLE;
#pragma unroll 4
    for (int j = 0; j < NUM_SAMPLE; ++j) {
        const int idx = nrow[j];          // uniform -> scalar load
        const float2 v =
            *(const float2*)(features + (size_t)idx * FEAT_DIM + 2 * t);
        acc.x += v.x;
        acc.y += v.y;
    }
    acc.x *= (1.0f / NUM_SAMPLE);
    acc.y *= (1.0f / NUM_SAMPLE);

    float* row = combined + (size_t)b * K_DIM;
    *(float2*)(row + 2 * t)            = selfv;  // self half
    *(float2*)(row + FEAT_DIM + 2 * t) = acc;    // neighbor-mean half
}

// ---------------------------------------------------------------------------
// Kernel 2: out[512,16384] = relu(W[512,1024] @ combined[16384,1024]^T)
//
// Block = 256 threads = 8 waves; block tile 128(M) x 128(N).
// Waves: wm = wave>>1 (4 M groups of 32 rows), wn = wave&1 (2 N groups of 64).
// Each wave: 2 M-subtiles x 4 N-subtiles of 16x16 -> 8 v8f accumulators.
//
// B tile (combined[n0..n0+127][kc..kc+15], 8 KB logical) is staged in LDS via
// GLOBAL_LOAD_ASYNC_TO_LDS_B128 (ASYNCcnt), double-buffered. Each of the 256
// threads copies one 16B segment per issue; 2 issues cover 128 rows.
// A fragments (weight) are loaded directly from global (2 MB, L2-resident).
//
// f32 WMMA fragment layouts (16x16x4):
//   A (16x4): lane l -> row m+(l&15),  K = k + 2*(l>>4) .. +1   (b64 global)
//   B (4x16): lane l -> col n+(l&15),  K = k + 2*(l>>4) .. +1   (b64 from LDS)
//   D: VGPR v, lane l -> row m + v + 8*(l>>4), col n + (l&15)
// ---------------------------------------------------------------------------
__global__ void __launch_bounds__(256)
gemm_relu_kernel(const float* __restrict__ W,
                 const float* __restrict__ Cmb,
                 float*       __restrict__ out) {
    __shared__ float smemB[2][BN * LDS_STRIDE];   // 2 x 10240 B

    const int t    = threadIdx.x;
    const int lane = t & 31;
    const int wave = t >> 5;       // 0..7
    const int wm   = wave >> 1;    // 0..3
    const int wn   = wave & 1;     // 0..1

    const int n0    = blockIdx.x * BN;
    const int mBase = blockIdx.y * BM + wm * 32;

    const int r  = lane & 15;
    const int kO = (lane >> 4) << 1;              // 0 or 2

    const float* aPtr0 = W + (size_t)(mBase +  0 + r) * K_DIM + kO;
    const float* aPtr1 = W + (size_t)(mBase + 16 + r) * K_DIM + kO;

    // Async-copy mapping: thread t covers row cr (and cr+64), 16B segment cs.
    const int cr = t >> 2;                        // 0..63
    const int cs = t & 3;                         // 0..3
    const float* gCopy =
        Cmb + (size_t)(n0 + cr) * K_DIM + cs * 4; // + kc per chunk
    // Raw LDS byte address = low 32 bits of the flat shared-memory pointer
    // (LDS aperture lives in addr[63:32]).
    const uint32_t lBase0 = (uint32_t)(uintptr_t)(&smemB[0][0]);
    const uint32_t lBase1 = (uint32_t)(uintptr_t)(&smemB[1][0]);
    const uint32_t lOff   =
        (uint32_t)((cr * LDS_STRIDE + cs * 4) * sizeof(float));
    const uint32_t lOff2  =
        lOff + (uint32_t)(64 * LDS_STRIDE * sizeof(float)); // rows +64

    v8f acc[2][4] = {};

    // Prologue: stage chunk 0 into buffer 0.
    {
        const uint64_t g0 = (uint64_t)(uintptr_t)(gCopy);
        const uint64_t g1 = (uint64_t)(uintptr_t)(gCopy + (size_t)64 * K_DIM);
        asm volatile("global_load_async_to_lds_b128 %0, %2, off\n\t"
                     "global_load_async_to_lds_b128 %1, %3, off"
                     :: "v"(lBase0 + lOff), "v"(lBase0 + lOff2),
                        "v"(g0), "v"(g1)
                     : "memory");
    }
    asm volatile("s_wait_asynccnt 0" ::: "memory");
    __syncthreads();

    for (int c = 0; c < NCHUNK; ++c) {
        const int kc = c * KC;

        // Issue async copy of next chunk into the other buffer.
        if (c + 1 < NCHUNK) {
            const uint32_t lb = ((c + 1) & 1) ? lBase1 : lBase0;
            const uint64_t g0 =
                (uint64_t)(uintptr_t)(gCopy + (kc + KC));
            const uint64_t g1 =
                (uint64_t)(uintptr_t)(gCopy + (size_t)64 * K_DIM + (kc + KC));
            asm volatile("global_load_async_to_lds_b128 %0, %2, off\n\t"
                         "global_load_async_to_lds_b128 %1, %3, off"
                         :: "v"(lb + lOff), "v"(lb + lOff2),
                            "v"(g0), "v"(g1)
                         : "memory");
        }

        // Compute on the current buffer.
        const float* sB =
            ((c & 1) ? smemB[1] : smemB[0]) + wn * 64 * LDS_STRIDE;
#pragma unroll
        for (int step = 0; step < 4; ++step) {
            const int kk = step * 4;
            const v2f a0 = *(const v2f*)(aPtr0 + kc + kk);
            const v2f a1 = *(const v2f*)(aPtr1 + kc + kk);
#pragma unroll
            for (int nt = 0; nt < 4; ++nt) {
                const v2f bf =
                    *(const v2f*)(sB + (nt * 16 + r) * LDS_STRIDE + kk + kO);
                acc[0][nt] = __builtin_amdgcn_wmma_f32_16x16x4_f32(
                    false, a0, false, bf, (short)0, acc[0][nt], false, false);
                acc[1][nt] = __builtin_amdgcn_wmma_f32_16x16x4_f32(
                    false, a1, false, bf, (short)0, acc[1][nt], false, false);
            }
        }

        // My async writes done; then block-wide barrier so (a) everyone's data
        // for chunk c+1 is visible and (b) buffer c&1 was fully consumed.
        asm volatile("s_wait_asynccnt 0" ::: "memory");
        __syncthreads();
    }

    // Fused ReLU + store. Lanes 0..15 write 16 consecutive f32 (64B) per row.
    const int rowOff = (lane >> 4) << 3;          // +8 for upper lane half
    const int col    = lane & 15;
#pragma unroll
    for (int sub = 0; sub < 2; ++sub) {
#pragma unroll
        for (int v = 0; v < 8; ++v) {
            const size_t row = (size_t)(mBase + sub * 16 + rowOff + v);
            float* o = out + row * BATCH + n0 + wn * 64 + col;
#pragma unroll
            for (int nt = 0; nt < 4; ++nt)
                o[nt * 16] = fmaxf(acc[sub][nt][v], 0.0f);
        }
    }
}

// ---------------------------------------------------------------------------
// Launch
// ---------------------------------------------------------------------------
extern "C" void kernel_launch(void* const* d_in, const int* in_sizes, int n_in,
                              void* d_out, int out_size, void* d_ws, size_t ws_size,
                              hipStream_t stream) {
    const float* features = (const float*)d_in[0];   // [100000, 512] f32
    const float* weight   = (const float*)d_in[1];   // [512, 1024]   f32
    const int*   nodes    = (const int*)d_in[2];     // [16384]       i32
    const int*   neigh    = (const int*)d_in[3];     // [16384, 32]   i32

    float* out      = (float*)d_out;                 // [512, 16384] f32
    float* combined = (float*)d_ws;                  // [16384, 1024] f32 (64 MB)

    // Stage 1: gather + mean + concat into workspace
    agg_combined_kernel<<<dim3(BATCH), dim3(256), 0, stream>>>(
        features, nodes, neigh, combined);

    // Stage 2: f32 WMMA GEMM with async-LDS B staging and fused ReLU
    // grid: (16384/128, 512/128) = (128, 4); exact division -> EXEC all-1s
    gemm_relu_kernel<<<dim3(BATCH / BN, EMBED_DIM / BM), dim3(256), 0, stream>>>(
        weight, combined, out);
}